// ConstrainedModel_84464826843702
// MI455X (gfx1250) — compile-verified
//
#include <hip/hip_runtime.h>

typedef float v2f __attribute__((ext_vector_type(2)));
typedef float v4f __attribute__((ext_vector_type(4)));
typedef float v8f __attribute__((ext_vector_type(8)));
typedef unsigned int v4u __attribute__((ext_vector_type(4)));
typedef int v4i __attribute__((ext_vector_type(4)));
typedef int v8i __attribute__((ext_vector_type(8)));

#define HD 100      // hidden width
#define LAYERS 4
#define KC 20       // K columns per LDS weight chunk
#define NKS 5       // K-steps (of 4) per chunk
#define NCHUNK 5    // chunks per layer (100/20)
#define NMT 7       // M tiles (112 = 7*16 rows, rows >=100 zero-padded)

#define MODE_RED 0  // w-reduced net eval (full_grid -> y_full)
#define MODE_PHI 1  // raw phi (ic_grid -> ic_pred)
#define MODE_BC  2  // value + d/dx, paired start/end -> bc_diff
#define MODE_RES 3  // value + d/dx + d/dt + d2/dx2 -> pde_residual

__device__ __forceinline__ float fast_tanh(float x) {
  float e = __expf(2.0f * x);
  return 1.0f - 2.0f / (e + 1.0f);
}

template<int NCH, int NWAVES, int MODE>
__global__ __launch_bounds__(NWAVES * 32)
void pinn_net_kernel(const float* __restrict__ grid0,
                     const float* __restrict__ grid1,   // bc end grid (MODE_BC only)
                     const float* __restrict__ w,       // (B,100)
                     const float* __restrict__ W0,      // (100,2)
                     const float* __restrict__ b0,      // (100)
                     const float* __restrict__ Ws,      // (4,100,100)
                     const float* __restrict__ bs,      // (4,100)
                     float* __restrict__ out)
{
  constexpr int NT = NWAVES * 32;
  constexpr int NTILE = NWAVES * 16;   // points per workgroup

  __shared__ float sS[NCH * HD * NTILE];   // activations, layout [ch][k][n]
  __shared__ float sW[112 * KC];           // weight K-chunk, layout [m][kc]
  __shared__ float sW0[2 * HD];
  __shared__ float sB0[HD];
  __shared__ float sBS[LAYERS * HD];
  __shared__ float sw[HD];
  __shared__ float sPX[NTILE];
  __shared__ float sPT[NTILE];

  const int tid  = threadIdx.x;
  const int lane = tid & 31;
  const int wave = tid >> 5;
  const int hi   = (lane >> 4) & 1;           // lane-half (wave32)
  const int nloc = wave * 16 + (lane & 15);   // this lane's point column

  // ---- stage small constants; zero weight pad rows (100..111) once ----
  for (int i = tid; i < 2 * HD; i += NT) sW0[i] = W0[i];
  for (int i = tid; i < HD; i += NT)     sB0[i] = b0[i];
  for (int i = tid; i < LAYERS * HD; i += NT) sBS[i] = bs[i];
  for (int i = tid; i < 12 * KC; i += NT) sW[HD * KC + i] = 0.0f;  // TDM never writes these

  // ---- stage point coords (+ batch weight vector for reduced modes) ----
  int batch = 0;
  if constexpr (MODE == MODE_RED) {
    long pbase = (long)blockIdx.x * NTILE;
    batch = (int)(pbase >> 14);               // 16384 full-grid pts / batch
    if (tid < NTILE) { sPX[tid] = grid0[(pbase + tid) * 2 + 0];
                       sPT[tid] = grid0[(pbase + tid) * 2 + 1]; }
  } else if constexpr (MODE == MODE_PHI) {
    long pbase = (long)blockIdx.x * NTILE;
    if (tid < NTILE) { sPX[tid] = grid0[(pbase + tid) * 2 + 0];
                       sPT[tid] = grid0[(pbase + tid) * 2 + 1]; }
  } else if constexpr (MODE == MODE_BC) {
    int pairbase = blockIdx.x * 32;           // 32 start + 32 end pts per WG
    if (tid < NTILE) {
      int j = pairbase + (tid & 31);
      const float* g = (tid < 32) ? grid0 : grid1;
      sPX[tid] = g[j * 2 + 0]; sPT[tid] = g[j * 2 + 1];
    }
  } else { // MODE_RES
    int pbase = blockIdx.x * NTILE;           // 32 residual pts per WG
    batch = pbase >> 11;                      // 2048 residual pts / batch
    if (tid < NTILE) {
      int p = pbase + tid;
      int b = p >> 11, n = p & 2047;
      long gi = (long)b * 4096 + 2048 + n;    // last 2048 of interior grid
      sPX[tid] = grid0[gi * 2 + 0]; sPT[tid] = grid0[gi * 2 + 1];
    }
  }
  if constexpr (MODE == MODE_RED || MODE == MODE_RES) {
    for (int i = tid; i < HD; i += NT) sw[i] = w[batch * HD + i];
  }
  __syncthreads();

  // ---- layer 0: S = transform(W0 p + b0), tangents u=W0*e constant ----
  for (int e = tid; e < HD * NTILE; e += NT) {
    int k = e / NTILE, n = e % NTILE;
    float wx = sW0[k * 2 + 0], wt = sW0[k * 2 + 1];
    float h = wx * sPX[n] + wt * sPT[n] + sB0[k];
    float s = fast_tanh(h);
    float d = 1.0f - s * s;
    sS[0 * HD * NTILE + k * NTILE + n] = s;
    if constexpr (NCH >= 2) sS[1 * HD * NTILE + k * NTILE + n] = d * wx;  // ux
    if constexpr (NCH >= 4) {
      sS[2 * HD * NTILE + k * NTILE + n] = d * wt;                        // ut
      sS[3 * HD * NTILE + k * NTILE + n] = -2.0f * s * d * wx * wx;       // uxx
    }
  }
  __syncthreads();

  // ---- TDM descriptor (constant part): 2-D tile 20 x 100 of f32,
  //      row stride 100 elements, DMA'd into sW rows 0..99 ----
  const unsigned sW_lds = (unsigned)(uintptr_t)&sW[0];
  v8i g1;
  g1[0] = 0x20000;                 // workgroup_mask=0, data_size=2 (4 bytes)
  g1[1] = (int)(20u << 16);        // tensor_dim0 = 20 (bits 63:48)
  g1[2] = (int)(100u << 16);       // tensor_dim1 = 100 (bits 95:80)
  g1[3] = (int)(20u << 16);        // tile_dim0 = 20 (bits 127:112)
  g1[4] = 100;                     // tile_dim1 = 100 (bits 143:128)
  g1[5] = 100;                     // tensor_dim0_stride = 100 (bits 207:160, low 32)
  g1[6] = 0;
  g1[7] = 0;
  const v4i z4 = {};
  const v8i z8 = {};

  v8f C[NCH][NMT];

  for (int L = 0; L < LAYERS; ++L) {
    // init accumulators: value channel gets bias, tangents zero
#pragma unroll
    for (int mt = 0; mt < NMT; ++mt) {
      v8f cz;
#pragma unroll
      for (int r = 0; r < 8; ++r) {
        int row = mt * 16 + r + 8 * hi;
        cz[r] = (row < HD) ? sBS[L * HD + row] : 0.0f;
      }
      C[0][mt] = cz;
#pragma unroll
      for (int ch = 1; ch < NCH; ++ch) { v8f z = {}; C[ch][mt] = z; }
    }

    for (int c = 0; c < NCHUNK; ++c) {
      __syncthreads();  // all waves done reading previous chunk's sW
      if (wave == 0) {
        // Tensor Data Mover: DMA W[:, c*KC .. c*KC+19] -> LDS (rows contiguous)
        unsigned long long ga =
            (unsigned long long)(uintptr_t)(Ws + (long)L * HD * HD + c * KC);
        v4u g0;
        g0[0] = 1u;                                   // count=1, user descriptor
        g0[1] = sW_lds;                               // lds_addr (bytes)
        g0[2] = (unsigned)(ga & 0xFFFFFFFFu);         // global_addr[31:0]
        g0[3] = (unsigned)((ga >> 32) & 0x01FFFFFFu)  // global_addr[56:32]
              | (2u << 30);                           // type = 2 ("image")
        __builtin_amdgcn_tensor_load_to_lds(g0, g1, z4, z4, z8, 0);
        __builtin_amdgcn_s_wait_tensorcnt(0);
      }
      __syncthreads();  // weights visible to all waves

      // B fragments for this chunk (own column; [k][n] LDS layout)
      float Br[NCH][NKS][2];
#pragma unroll
      for (int ch = 0; ch < NCH; ++ch)
#pragma unroll
        for (int ks = 0; ks < NKS; ++ks) {
          int ka = (c * NKS + ks) * 4 + 2 * hi;
          Br[ch][ks][0] = sS[ch * HD * NTILE + ka * NTILE + nloc];
          Br[ch][ks][1] = sS[ch * HD * NTILE + (ka + 1) * NTILE + nloc];
        }

#pragma unroll
      for (int mt = 0; mt < NMT; ++mt) {
        v2f A[NKS];
        int m = mt * 16 + (lane & 15);
#pragma unroll
        for (int ks = 0; ks < NKS; ++ks)
          A[ks] = *(const v2f*)(sW + m * KC + ks * 4 + 2 * hi);
#pragma unroll
        for (int ch = 0; ch < NCH; ++ch)
#pragma unroll
          for (int ks = 0; ks < NKS; ++ks) {
            v2f b2 = { Br[ch][ks][0], Br[ch][ks][1] };
            C[ch][mt] = __builtin_amdgcn_wmma_f32_16x16x4_f32(
                false, A[ks], false, b2, (short)0, C[ch][mt], false, false);
          }
      }
    }
    __syncthreads();

    if (L < LAYERS - 1) {
      // tanh transform of own 16 columns, back into sS
#pragma unroll
      for (int mt = 0; mt < NMT; ++mt)
#pragma unroll
        for (int r = 0; r < 8; ++r) {
          int row = mt * 16 + r + 8 * hi;
          if (row >= HD) continue;
          float h = C[0][mt][r];
          float s = fast_tanh(h);
          float d = 1.0f - s * s;
          sS[0 * HD * NTILE + row * NTILE + nloc] = s;
          if constexpr (NCH >= 2) {
            float ux = C[1][mt][r];
            sS[1 * HD * NTILE + row * NTILE + nloc] = d * ux;
            if constexpr (NCH >= 4) {
              float ut = C[2][mt][r], uxx = C[3][mt][r];
              sS[2 * HD * NTILE + row * NTILE + nloc] = d * ut;
              sS[3 * HD * NTILE + row * NTILE + nloc] = d * uxx - 2.0f * s * d * ux * ux;
            }
          }
        }
      __syncthreads();
    }
  }

  // ---- epilogue ----
  if constexpr (MODE == MODE_PHI) {
    long p = (long)blockIdx.x * NTILE + nloc;
    float* o = out + p * HD;
#pragma unroll
    for (int mt = 0; mt < NMT; ++mt)
#pragma unroll
      for (int r = 0; r < 8; ++r) {
        int row = mt * 16 + r + 8 * hi;
        if (row < HD) o[row] = C[0][mt][r];
      }
  } else if constexpr (MODE == MODE_RED || MODE == MODE_RES) {
    float acc[NCH];
#pragma unroll
    for (int ch = 0; ch < NCH; ++ch) acc[ch] = 0.0f;
#pragma unroll
    for (int mt = 0; mt < NMT; ++mt)
#pragma unroll
      for (int r = 0; r < 8; ++r) {
        int row = mt * 16 + r + 8 * hi;
        if (row < HD) {
          float wv = sw[row];
#pragma unroll
          for (int ch = 0; ch < NCH; ++ch) acc[ch] += wv * C[ch][mt][r];
        }
      }
#pragma unroll
    for (int ch = 0; ch < NCH; ++ch) acc[ch] += __shfl_xor(acc[ch], 16, 32);
    if (lane < 16) {
      long p = (long)blockIdx.x * NTILE + nloc;
      if constexpr (MODE == MODE_RED) {
        out[p] = acc[0];
      } else {
        out[p] = acc[2] + acc[0] * acc[1] - 0.01f * acc[3];  // dy_t + y*dy_x - nu*dy_xx
      }
    }
  } else { // MODE_BC
    if (wave >= 2) {  // end-point waves publish fragments via sS
      int n32 = nloc - 32;
#pragma unroll
      for (int mt = 0; mt < NMT; ++mt)
#pragma unroll
        for (int r = 0; r < 8; ++r) {
          int row = mt * 16 + r + 8 * hi;
          if (row < HD) {
            sS[0 * HD * NTILE + row * NTILE + n32] = C[0][mt][r];
            sS[(NCH > 1 ? 1 : 0) * HD * NTILE + row * NTILE + n32] = C[NCH > 1 ? 1 : 0][mt][r];
          }
        }
    }
    __syncthreads();
    if (wave < 2) {   // start-point waves subtract and store bc_diff
      int n32 = nloc;
      long p = (long)blockIdx.x * 32 + n32;      // == b*128 + t
      float* o = out + p * 200;
#pragma unroll
      for (int mt = 0; mt < NMT; ++mt)
#pragma unroll
        for (int r = 0; r < 8; ++r) {
          int row = mt * 16 + r + 8 * hi;
          if (row < HD) {
            o[row]        = C[0][mt][r] - sS[0 * HD * NTILE + row * NTILE + n32];
            o[100 + row]  = C[NCH > 1 ? 1 : 0][mt][r]
                          - sS[(NCH > 1 ? 1 : 0) * HD * NTILE + row * NTILE + n32];
          }
        }
    }
  }
}

extern "C" void kernel_launch(void* const* d_in, const int* in_sizes, int n_in,
                              void* d_out, int out_size, void* d_ws, size_t ws_size,
                              hipStream_t stream) {
  (void)in_sizes; (void)n_in; (void)out_size; (void)d_ws; (void)ws_size;
  // setup_inputs order: 0 ic_vals (unused by reference), 1 ic_grid, 2 bc_grid_start,
  // 3 bc_grid_end, 4 interior_grid, 5 full_grid, 6 w, 7 W0, 8 b0, 9 Ws, 10 bs
  const float* ic_grid  = (const float*)d_in[1];
  const float* bc_s     = (const float*)d_in[2];
  const float* bc_e     = (const float*)d_in[3];
  const float* interior = (const float*)d_in[4];
  const float* full     = (const float*)d_in[5];
  const float* w        = (const float*)d_in[6];
  const float* W0       = (const float*)d_in[7];
  const float* b0       = (const float*)d_in[8];
  const float* Ws       = (const float*)d_in[9];
  const float* bs       = (const float*)d_in[10];

  float* out = (float*)d_out;
  float* y_full  = out;                               // 16*16384   = 262144
  float* ic_pred = out + 262144;                      // 16*128*100 = 204800
  float* bc_diff = out + 262144 + 204800;             // 16*128*200 = 409600
  float* pde_res = out + 262144 + 204800 + 409600;    // 16*2048    = 32768

  // y_full: 262144 points, 128/WG, 8 waves
  pinn_net_kernel<1, 8, MODE_RED><<<dim3(2048), dim3(256), 0, stream>>>(
      full, nullptr, w, W0, b0, Ws, bs, y_full);
  // ic_pred: 2048 points
  pinn_net_kernel<1, 8, MODE_PHI><<<dim3(16), dim3(256), 0, stream>>>(
      ic_grid, nullptr, w, W0, b0, Ws, bs, ic_pred);
  // bc_diff: 2048 pairs, 32 pairs/WG (waves 0-1 start, 2-3 end)
  pinn_net_kernel<2, 4, MODE_BC><<<dim3(64), dim3(128), 0, stream>>>(
      bc_s, bc_e, w, W0, b0, Ws, bs, bc_diff);
  // pde_residual: 32768 points, 32/WG, 4 derivative channels
  pinn_net_kernel<4, 2, MODE_RES><<<dim3(1024), dim3(64), 0, stream>>>(
      interior, nullptr, w, W0, b0, Ws, bs, pde_res);
}